// CNN_LSTM_Attention_Classifier_89395449299804
// MI455X (gfx1250) — compile-verified
//
#include <hip/hip_runtime.h>
#include <math.h>

typedef __attribute__((ext_vector_type(2))) float v2f;
typedef __attribute__((ext_vector_type(8))) float v8f;

#define B_ 128
#define T_ 129
#define H_ 256
#define G_ 1024
#define BN_SCALE_F 0.999995000037499f   // 1/sqrt(1+1e-5)

// ---------------------------------------------------------------------------
// Fused conv1d + bias + maxpool(PK) + BN*scale + ReLU   (fp32, weights in LDS)
// in:  (B, CIN, Lin)   w: (Cout, CIN, K)   out: (B, Cout, Lout)
// ---------------------------------------------------------------------------
template<int CIN, int K, int P, int PK, bool ACT>
__global__ void conv_pool(const float* __restrict__ in, int Lin,
                          const float* __restrict__ w, const float* __restrict__ bias,
                          float* __restrict__ out, int Lout, int Cout)
{
  __shared__ float wsm[CIN * K];
  const int co = blockIdx.y;
  const int b  = blockIdx.z;
  for (int i = threadIdx.x; i < CIN * K; i += blockDim.x)
    wsm[i] = w[co * CIN * K + i];
  __syncthreads();
  const int lo = blockIdx.x * blockDim.x + threadIdx.x;
  if (lo >= Lout) return;
  const float bv = bias[co];
  const float* inb = in + (long)b * CIN * Lin;
  float m = -__builtin_inff();
  for (int j = 0; j < PK; ++j) {
    const int l = lo * PK + j;
    float acc = bv;
    for (int ci = 0; ci < CIN; ++ci) {
      const float* row = inb + ci * Lin;
      const float* wr  = wsm + ci * K;
      #pragma unroll
      for (int k = 0; k < K; ++k) {
        const int li = l - P + k;
        const float xv = (li >= 0 && li < Lin) ? row[li] : 0.f;
        acc = fmaf(xv, wr[k], acc);
      }
    }
    m = fmaxf(m, acc);
  }
  if (ACT) m = fmaxf(m * BN_SCALE_F, 0.f);
  out[((long)b * Cout + co) * Lout + lo] = m;
}

// ---------------------------------------------------------------------------
// WMMA fp32 GEMM, compile-time K segments, NT=8 register blocking, ping-pong
// double-buffered operand fetch (no inter-buffer copies):
//   C(M,N) = [A0|A1|A2](M, K0+K1+K2) * Bw(Ktot, N)
// One wave (32 threads) computes a 16 x (16*NT) strip of C.
// grid = (M/16, N/(16*NT)), block = 32. Layouts per CDNA5 ISA 7.12.2.
// ---------------------------------------------------------------------------
template<int KSEG, int NT>
__device__ __forceinline__ void gemm_seg(const float* __restrict__ ap,
                                         const float* __restrict__& bp,
                                         const int ldb, v8f (&acc)[NT])
{
  constexpr int STEPS = KSEG / 4;   // even for all uses (K multiples of 8)
  v2f a0, a1;
  v2f b0[NT], b1[NT];

  // prologue: step 0 -> buffer 0
  a0.x = ap[0]; a0.y = ap[1];                   // A[m, k+2h], A[m, k+2h+1]
  #pragma unroll
  for (int nt = 0; nt < NT; ++nt) {             // B[k+2h, n], B[k+2h+1, n]
    b0[nt].x = bp[nt * 16];
    b0[nt].y = bp[ldb + nt * 16];
  }

  for (int i = 0; i < STEPS; i += 2) {
    // prefetch step i+1 -> buffer 1
    {
      const float* a_p = ap + 4;
      const float* b_p = bp + 4 * ldb;
      a1.x = a_p[0]; a1.y = a_p[1];
      #pragma unroll
      for (int nt = 0; nt < NT; ++nt) {
        b1[nt].x = b_p[nt * 16];
        b1[nt].y = b_p[ldb + nt * 16];
      }
    }
    // compute step i from buffer 0
    #pragma unroll
    for (int nt = 0; nt < NT; ++nt)
      acc[nt] = __builtin_amdgcn_wmma_f32_16x16x4_f32(false, a0, false, b0[nt],
                                                      (short)0, acc[nt], false, false);
    // prefetch step i+2 -> buffer 0
    if (i + 2 < STEPS) {
      const float* a_p = ap + 8;
      const float* b_p = bp + 8 * ldb;
      a0.x = a_p[0]; a0.y = a_p[1];
      #pragma unroll
      for (int nt = 0; nt < NT; ++nt) {
        b0[nt].x = b_p[nt * 16];
        b0[nt].y = b_p[ldb + nt * 16];
      }
    }
    // compute step i+1 from buffer 1
    #pragma unroll
    for (int nt = 0; nt < NT; ++nt)
      acc[nt] = __builtin_amdgcn_wmma_f32_16x16x4_f32(false, a1, false, b1[nt],
                                                      (short)0, acc[nt], false, false);
    ap += 8;
    bp += 8 * ldb;
  }
}

template<int K0, int K1, int K2, int NT>
__global__ void gemm_wmma(const float* __restrict__ A0, int lda0,
                          const float* __restrict__ A1, int lda1,
                          const float* __restrict__ A2, int lda2,
                          const float* __restrict__ Bw, int ldb,
                          float* __restrict__ C, int ldc)
{
  const int lane = threadIdx.x;
  const int half = lane >> 4;       // 0: K+0/K+1 ; 1: K+2/K+3
  const int lm   = lane & 15;
  const int m0   = blockIdx.x * 16;
  const int n0   = blockIdx.y * (16 * NT);
  const int row  = m0 + lm;

  v8f acc[NT];
  #pragma unroll
  for (int nt = 0; nt < NT; ++nt) {
    v8f z = {0.f, 0.f, 0.f, 0.f, 0.f, 0.f, 0.f, 0.f};
    acc[nt] = z;
  }

  const float* bp = Bw + (long)(2 * half) * ldb + n0 + lm;

  gemm_seg<K0, NT>(A0 + (long)row * lda0 + 2 * half, bp, ldb, acc);
  if constexpr (K1 > 0)
    gemm_seg<K1, NT>(A1 + (long)row * lda1 + 2 * half, bp, ldb, acc);
  if constexpr (K2 > 0)
    gemm_seg<K2, NT>(A2 + (long)row * lda2 + 2 * half, bp, ldb, acc);

  #pragma unroll
  for (int nt = 0; nt < NT; ++nt) {
    #pragma unroll
    for (int r = 0; r < 8; ++r)
      C[(long)(m0 + r + half * 8) * ldc + n0 + nt * 16 + lm] = acc[nt][r];
  }
}

// ---------------------------------------------------------------------------
// LSTM pointwise: gates g (B,1024) (+bih+bhh) -> update h,c; optionally store
// h into ann[:, t, :].
// ---------------------------------------------------------------------------
__global__ void lstm_update(const float* __restrict__ g,
                            const float* __restrict__ bih, const float* __restrict__ bhh,
                            float* __restrict__ h, float* __restrict__ c,
                            float* __restrict__ ann_out, int t)
{
  const int idx = blockIdx.x * blockDim.x + threadIdx.x;
  if (idx >= B_ * H_) return;
  const int b = idx >> 8, j = idx & 255;
  const float* gb = g + b * G_;
  const float gi = gb[j]        + bih[j]        + bhh[j];
  const float gf = gb[256 + j]  + bih[256 + j]  + bhh[256 + j];
  const float gg = gb[512 + j]  + bih[512 + j]  + bhh[512 + j];
  const float go = gb[768 + j]  + bih[768 + j]  + bhh[768 + j];
  const float si = 1.f / (1.f + expf(-gi));
  const float sf = 1.f / (1.f + expf(-gf));
  const float so = 1.f / (1.f + expf(-go));
  const float cn = sf * c[idx] + si * tanhf(gg);
  const float hn = so * tanhf(cn);
  c[idx] = cn; h[idx] = hn;
  if (ann_out) ann_out[((long)b * T_ + t) * H_ + j] = hn;
}

// ---------------------------------------------------------------------------
// Attention score: sc[b,t] = a_b2 + sum_h a_w2[h]*relu(q[b,h]+annp[b,t,h]+a_b1[h])
// One 256-thread block per (t,b); LDS tree reduction.
// ---------------------------------------------------------------------------
__global__ void attn_scores(const float* __restrict__ q, const float* __restrict__ annp,
                            const float* __restrict__ a_b1, const float* __restrict__ a_w2,
                            const float* __restrict__ a_b2, float* __restrict__ sc)
{
  const int t = blockIdx.x, b = blockIdx.y, h = threadIdx.x;
  __shared__ float red[256];
  const float u = fmaxf(q[b * H_ + h] + annp[((long)b * T_ + t) * H_ + h] + a_b1[h], 0.f);
  red[h] = u * a_w2[h];
  __syncthreads();
  for (int s = 128; s > 0; s >>= 1) {
    if (h < s) red[h] += red[h + s];
    __syncthreads();
  }
  if (h == 0) sc[b * T_ + t] = red[0] + a_b2[0];
}

// ---------------------------------------------------------------------------
// Softmax over T + context: ctx[b,h] = sum_t softmax(sc[b,:])[t] * ann[b,t,h]
// One 256-thread block per batch row.
// ---------------------------------------------------------------------------
__global__ void attn_ctx(const float* __restrict__ sc, const float* __restrict__ ann,
                         float* __restrict__ ctx)
{
  const int b = blockIdx.x, tid = threadIdx.x;
  __shared__ float wgt[T_];
  __shared__ float red[256];
  const float v = (tid < T_) ? sc[b * T_ + tid] : -__builtin_inff();
  red[tid] = v; __syncthreads();
  for (int s = 128; s > 0; s >>= 1) {
    if (tid < s) red[tid] = fmaxf(red[tid], red[tid + s]);
    __syncthreads();
  }
  const float mx = red[0]; __syncthreads();
  const float e = (tid < T_) ? expf(v - mx) : 0.f;
  if (tid < T_) wgt[tid] = e;
  red[tid] = e; __syncthreads();
  for (int s = 128; s > 0; s >>= 1) {
    if (tid < s) red[tid] += red[tid + s];
    __syncthreads();
  }
  const float inv = 1.f / red[0]; __syncthreads();
  float acc = 0.f;
  for (int t = 0; t < T_; ++t)
    acc = fmaf(wgt[t], ann[((long)b * T_ + t) * H_ + tid], acc);
  ctx[b * H_ + tid] = acc * inv;
}

// ---------------------------------------------------------------------------
// Final head: out = sigmoid(relu(h@f_w1^T + f_b1) @ f_w2^T + f_b2)   (128,8)
// ---------------------------------------------------------------------------
__global__ void head(const float* __restrict__ h,
                     const float* __restrict__ fw1, const float* __restrict__ fb1,
                     const float* __restrict__ fw2, const float* __restrict__ fb2,
                     float* __restrict__ out)
{
  const int b = blockIdx.x * blockDim.x + threadIdx.x;
  if (b >= B_) return;
  float y[16];
  #pragma unroll
  for (int j = 0; j < 16; ++j) {
    float a = fb1[j];
    for (int k = 0; k < H_; ++k) a = fmaf(h[b * H_ + k], fw1[j * H_ + k], a);
    y[j] = fmaxf(a, 0.f);
  }
  #pragma unroll
  for (int o = 0; o < 8; ++o) {
    float a = fb2[o];
    #pragma unroll
    for (int k = 0; k < 16; ++k) a = fmaf(y[k], fw2[o * 16 + k], a);
    out[b * 8 + o] = 1.f / (1.f + expf(-a));
  }
}

// ---------------------------------------------------------------------------
// Weight prep: W[(k)*N + j] = k<K1 ? wih[j,k] : whh[j,k-K1]   (transpose+concat)
// ---------------------------------------------------------------------------
__global__ void build_wcat(const float* __restrict__ wih, const float* __restrict__ whh,
                           float* __restrict__ W, int K1, int K2, int N)
{
  const long total = (long)(K1 + K2) * N;
  const long i = (long)blockIdx.x * blockDim.x + threadIdx.x;
  if (i >= total) return;
  const int k = (int)(i / N), j = (int)(i % N);
  W[i] = (k < K1) ? wih[(long)j * K1 + k] : whh[(long)j * K2 + (k - K1)];
}

// a_w1 (256,512) -> Bq[k*256+j]=a_w1[j,k] (k<256), Ba[k*256+j]=a_w1[j,256+k]
__global__ void build_attn_w(const float* __restrict__ a_w1,
                             float* __restrict__ Bq, float* __restrict__ Ba)
{
  const int i = blockIdx.x * blockDim.x + threadIdx.x;
  if (i >= 256 * 256) return;
  const int k = i >> 8, j = i & 255;
  Bq[i] = a_w1[j * 512 + k];
  Ba[i] = a_w1[j * 512 + 256 + k];
}

__global__ void zero_f(float* __restrict__ p, int n)
{
  const int i = blockIdx.x * blockDim.x + threadIdx.x;
  if (i < n) p[i] = 0.f;
}

// ---------------------------------------------------------------------------
extern "C" void kernel_launch(void* const* d_in, const int* in_sizes, int n_in,
                              void* d_out, int out_size, void* d_ws, size_t ws_size,
                              hipStream_t stream)
{
  const float* x     = (const float*)d_in[0];
  const float* e_w1  = (const float*)d_in[1];  const float* e_b1 = (const float*)d_in[2];
  const float* e_w2  = (const float*)d_in[3];  const float* e_b2 = (const float*)d_in[4];
  const float* e_w3  = (const float*)d_in[5];  const float* e_b3 = (const float*)d_in[6];
  const float* e_w5  = (const float*)d_in[7];  const float* e_b5 = (const float*)d_in[8];
  const float* d_w1  = (const float*)d_in[9];  const float* d_b1 = (const float*)d_in[10];
  const float* d_w2  = (const float*)d_in[11]; const float* d_b2 = (const float*)d_in[12];
  const float* d_w3  = (const float*)d_in[13]; const float* d_b3 = (const float*)d_in[14];
  const float* d_w5  = (const float*)d_in[15]; const float* d_b5 = (const float*)d_in[16];
  const float* e_wih = (const float*)d_in[17]; const float* e_whh = (const float*)d_in[18];
  const float* e_bih = (const float*)d_in[19]; const float* e_bhh = (const float*)d_in[20];
  const float* d_wih = (const float*)d_in[21]; const float* d_whh = (const float*)d_in[22];
  const float* d_bih = (const float*)d_in[23]; const float* d_bhh = (const float*)d_in[24];
  const float* a_w1  = (const float*)d_in[25]; const float* a_b1 = (const float*)d_in[26];
  const float* a_w2  = (const float*)d_in[27]; const float* a_b2 = (const float*)d_in[28];
  const float* f_w1  = (const float*)d_in[29]; const float* f_b1 = (const float*)d_in[30];
  const float* f_w2  = (const float*)d_in[31]; const float* f_b2 = (const float*)d_in[32];
  float* out = (float*)d_out;
  float* ws  = (float*)d_ws;

  // workspace layout (float offsets); annp aliases bufA (free after convnets)
  const long o_bufA = 0;                               // 128*64*1023  = 8,380,416
  const long o_bufB = o_bufA + 8380416L;               // 128*128*255  = 4,177,920
  const long o_bufC = o_bufB + 4177920L;               // 128*256*128  = 4,194,304
  const long o_seq  = o_bufC + 4194304L;               // 128*129*256  = 4,227,072
  const long o_ann  = o_seq  + 4227072L;               // 128*129*256
  const long o_We   = o_ann  + 4227072L;               // 512*1024
  const long o_Wd   = o_We   + 524288L;                // 768*1024
  const long o_Bq   = o_Wd   + 786432L;                // 256*256
  const long o_Ba   = o_Bq   + 65536L;                 // 256*256
  const long o_h    = o_Ba   + 65536L;                 // 128*256
  const long o_c    = o_h    + 32768L;                 // 128*256
  const long o_g    = o_c    + 32768L;                 // 128*1024
  const long o_q    = o_g    + 131072L;                // 128*256
  const long o_sc   = o_q    + 32768L;                 // 128*129 (pad 16640)
  const long o_ctx  = o_sc   + 16640L;                 // 128*256
  float* bufA = ws + o_bufA; float* bufB = ws + o_bufB; float* bufC = ws + o_bufC;
  float* seq  = ws + o_seq;  float* ann  = ws + o_ann;  float* annp = ws + o_bufA;
  float* We   = ws + o_We;   float* Wd   = ws + o_Wd;
  float* Bq   = ws + o_Bq;   float* Ba   = ws + o_Ba;
  float* h    = ws + o_h;    float* c    = ws + o_c;    float* g   = ws + o_g;
  float* q    = ws + o_q;    float* sc   = ws + o_sc;   float* ctx = ws + o_ctx;

  // --- weight prep + state init ---
  build_wcat<<<dim3((512 * 1024 + 255) / 256), 256, 0, stream>>>(e_wih, e_whh, We, 256, 256, 1024);
  build_wcat<<<dim3((768 * 1024 + 255) / 256), 256, 0, stream>>>(d_wih, d_whh, Wd, 512, 256, 1024);
  build_attn_w<<<dim3(256), 256, 0, stream>>>(a_w1, Bq, Ba);
  zero_f<<<dim3(256), 256, 0, stream>>>(h, 65536);   // zeros h AND c (contiguous)

  // --- encoder convnet: x(raw-reshaped B,64,4096) -> seq (B,129,256 raw) ---
  conv_pool<64, 7, 2, 4, true ><<<dim3(4, 64, B_),  256, 0, stream>>>(x,    4096, e_w1, e_b1, bufA, 1023, 64);
  conv_pool<64, 5, 2, 4, true ><<<dim3(1, 128, B_), 256, 0, stream>>>(bufA, 1023, e_w2, e_b2, bufB, 255, 128);
  conv_pool<128,3, 2, 2, true ><<<dim3(1, 256, B_), 256, 0, stream>>>(bufB, 255,  e_w3, e_b3, bufC, 128, 256);
  conv_pool<256,4, 2, 1, false><<<dim3(1, 256, B_), 256, 0, stream>>>(bufC, 128,  e_w5, e_b5, seq,  129, 256);

  // --- encoder LSTM scan (T=129): gates via WMMA GEMM, pointwise update ---
  for (int t = 0; t < T_; ++t) {
    gemm_wmma<256, 256, 0, 8><<<dim3(B_ / 16, G_ / 128), 32, 0, stream>>>(
        seq + t * H_, T_ * H_,  h, H_,  h, H_,  We, G_, g, G_);
    lstm_update<<<dim3(B_), 256, 0, stream>>>(g, e_bih, e_bhh, h, c, ann, t);
  }

  // --- decoder convnet (reuses bufA/B/C, overwrites seq) ---
  conv_pool<64, 7, 2, 4, true ><<<dim3(4, 64, B_),  256, 0, stream>>>(x,    4096, d_w1, d_b1, bufA, 1023, 64);
  conv_pool<64, 5, 2, 4, true ><<<dim3(1, 128, B_), 256, 0, stream>>>(bufA, 1023, d_w2, d_b2, bufB, 255, 128);
  conv_pool<128,3, 2, 2, true ><<<dim3(1, 256, B_), 256, 0, stream>>>(bufB, 255,  d_w3, d_b3, bufC, 128, 256);
  conv_pool<256,4, 2, 1, false><<<dim3(1, 256, B_), 256, 0, stream>>>(bufC, 128,  d_w5, d_b5, seq,  129, 256);

  // --- hoisted attention key projection: annp = ann @ W_ann^T (WMMA) ---
  gemm_wmma<256, 0, 0, 8><<<dim3((B_ * T_) / 16, H_ / 128), 32, 0, stream>>>(
      ann, H_,  ann, H_,  ann, H_,  Ba, H_, annp, H_);

  // --- decoder scan with attention ---
  for (int t = 0; t < T_; ++t) {
    gemm_wmma<256, 0, 0, 8><<<dim3(B_ / 16, H_ / 128), 32, 0, stream>>>(   // q = h @ W_q^T
        h, H_,  h, H_,  h, H_,  Bq, H_, q, H_);
    attn_scores<<<dim3(T_, B_), 256, 0, stream>>>(q, annp, a_b1, a_w2, a_b2, sc);
    attn_ctx<<<dim3(B_), 256, 0, stream>>>(sc, ann, ctx);
    gemm_wmma<256, 256, 256, 8><<<dim3(B_ / 16, G_ / 128), 32, 0, stream>>>(  // gates: [xt|ctx|h] @ Wd
        seq + t * H_, T_ * H_,  ctx, H_,  h, H_,  Wd, G_, g, G_);
    lstm_update<<<dim3(B_), 256, 0, stream>>>(g, d_bih, d_bhh, h, c, nullptr, 0);
  }

  // --- classifier head ---
  head<<<dim3(1), 128, 0, stream>>>(h, f_w1, f_b1, f_w2, f_b2, out);
}